// Intergrator_5952824672851
// MI455X (gfx1250) — compile-verified
//
#include <hip/hip_runtime.h>
#include <hip/hip_bf16.h>

// FVM cell integrator for MI455X (gfx1250).
//
// Memory-bound kernel (0.35 FLOP/byte): WMMA is inapplicable (per-cell 2-elem
// dot products, no shared operand matrix). Optimization strategy:
//  * Keep the 144MB gathered face working set resident in the 192MB L2 by
//    marking everything touched exactly once (indices, rho, rhs_coef, unv,
//    outputs) non-temporal.
//  * Stage the per-cell unv (24B/cell) through LDS with the gfx1250 async
//    global->LDS path (ASYNCcnt, th:TH_LOAD_NT), issued before the dependent
//    index-gather->face-gather chain; the s_wait_asynccnt takes the gathered
//    values as asm inputs so the compiler cannot hoist it above the gathers
//    (observed in round 2: wait was placed right after the async issues,
//    killing the overlap).

#define THREADS 256

// clang-native 2-float vector: accepted by __builtin_nontemporal_load/store
// (HIP's float2 is a struct and is rejected by those builtins).
typedef float v2f __attribute__((ext_vector_type(2)));

__global__ __launch_bounds__(THREADS) void fvm_cell_kernel(
    const v2f*   __restrict__ uv_face,    // [F]
    const float* __restrict__ p_face,     // [F]
    const v2f*   __restrict__ flux_D,     // [F]
    const float* __restrict__ unv,        // [C*6]
    const float* __restrict__ rho,        // [C]
    const float* __restrict__ rhs_coef,   // [C]
    const float* __restrict__ face_area,  // [F]
    const int*   __restrict__ cell_face,  // [3*C]
    float*       __restrict__ out_loss,   // [C]
    v2f*         __restrict__ out_rhs,    // [C]
    int C)
{
    __shared__ __align__(16) float s_unv[THREADS * 6];

    const int tid = (int)threadIdx.x;
    const long long c  = (long long)blockIdx.x * THREADS + tid;
    const long long cc = (c < (long long)C) ? c : (long long)(C - 1);  // clamp tail lanes

    // ---- async-stage this cell's 3 unit normals (24B) into LDS ----------
    // Issued first so the ASYNC transfer overlaps the dependent gather chain
    // below. NT hint: unv is a 96MB one-shot stream; don't let it evict the
    // L2-resident face arrays. Each lane consumes only its own 24B slice ->
    // the per-wave s_wait_asynccnt is the only sync needed, no barrier.
    {
        unsigned lds_addr =
            (unsigned)(unsigned long long)(&s_unv[tid * 6]);   // low 32 bits = LDS offset
        unsigned long long gaddr =
            (unsigned long long)(const void*)(unv + 6ll * cc); // 8B-aligned (24B stride)
        asm volatile(
            "global_load_async_to_lds_b64 %0, %1, off th:TH_LOAD_NT\n\t"
            "global_load_async_to_lds_b64 %0, %1, off offset:8 th:TH_LOAD_NT\n\t"
            "global_load_async_to_lds_b64 %0, %1, off offset:16 th:TH_LOAD_NT"
            :: "v"(lds_addr), "v"(gaddr)
            : "memory");
    }

    // ---- face indices (streamed once -> non-temporal) -------------------
    const int f0 = __builtin_nontemporal_load(cell_face + cc);
    const int f1 = __builtin_nontemporal_load(cell_face + (long long)C + cc);
    const int f2 = __builtin_nontemporal_load(cell_face + 2ll * (long long)C + cc);

    // ---- gathered face data: default (RT) policy so it stays L2-resident -
    const v2f   uv0 = uv_face[f0];
    const v2f   uv1 = uv_face[f1];
    const v2f   uv2 = uv_face[f2];
    const float e0  = face_area[f0];
    const float e1  = face_area[f1];
    const float e2  = face_area[f2];
    const float p0  = p_face[f0];
    const float p1  = p_face[f1];
    const float p2  = p_face[f2];
    const v2f   fd0 = flux_D[f0];
    const v2f   fd1 = flux_D[f1];
    const v2f   fd2 = flux_D[f2];

    // ---- per-cell streams (non-temporal) --------------------------------
    const float r  = __builtin_nontemporal_load(rho + cc);
    const float rc = __builtin_nontemporal_load(rhs_coef + cc);

    // ---- consume staged unv from LDS ------------------------------------
    // The dummy "v" inputs tie this wait AFTER the gather loads: the async
    // transfer overlaps the whole index->gather chain and is complete by the
    // time we get here.
    asm volatile("s_wait_asynccnt 0x0"
                 :: "v"(uv0.x), "v"(uv1.x), "v"(uv2.x), "v"(fd2.x), "v"(e2), "v"(rc)
                 : "memory");
    const float n0x = s_unv[tid * 6 + 0];
    const float n0y = s_unv[tid * 6 + 1];
    const float n1x = s_unv[tid * 6 + 2];
    const float n1y = s_unv[tid * 6 + 3];
    const float n2x = s_unv[tid * 6 + 4];
    const float n2y = s_unv[tid * 6 + 5];

    // ---- math -----------------------------------------------------------
    // s_i = dot(uv_i, n_i) * A_i ; loss = sum s_i ; fluxA = sum uv_i * s_i
    const float s0 = fmaf(uv0.x, n0x, uv0.y * n0y) * e0;
    const float s1 = fmaf(uv1.x, n1x, uv1.y * n1y) * e1;
    const float s2 = fmaf(uv2.x, n2x, uv2.y * n2y) * e2;

    const float lc = s0 + s1 + s2;

    const float Ax = fmaf(uv2.x, s2, fmaf(uv1.x, s1, uv0.x * s0));
    const float Ay = fmaf(uv2.y, s2, fmaf(uv1.y, s1, uv0.y * s0));

    const float pe0 = p0 * e0, pe1 = p1 * e1, pe2 = p2 * e2;
    const float Px = fmaf(pe2, n2x, fmaf(pe1, n1x, pe0 * n0x));
    const float Py = fmaf(pe2, n2y, fmaf(pe1, n1y, pe0 * n0y));

    const float Dx = fd0.x + fd1.x + fd2.x;
    const float Dy = fd0.y + fd1.y + fd2.y;

    const float invr = 1.0f / r;
    const float ox = fmaf(rc, -Ax - invr * Px, Dx);
    const float oy = fmaf(rc, -Ay - invr * Py, Dy);

    // ---- non-temporal stores (don't pollute L2) -------------------------
    if (c < (long long)C) {
        __builtin_nontemporal_store(lc, out_loss + c);
        v2f o;
        o.x = ox;
        o.y = oy;
        __builtin_nontemporal_store(o, out_rhs + c);
    }
}

extern "C" void kernel_launch(void* const* d_in, const int* in_sizes, int n_in,
                              void* d_out, int out_size, void* d_ws, size_t ws_size,
                              hipStream_t stream) {
    const v2f*   uv_face   = (const v2f*)d_in[0];
    const float* p_face    = (const float*)d_in[1];
    const v2f*   flux_D    = (const v2f*)d_in[2];
    const float* unv       = (const float*)d_in[3];
    const float* rho       = (const float*)d_in[4];
    const float* rhs_coef  = (const float*)d_in[5];
    const float* face_area = (const float*)d_in[6];
    const int*   cell_face = (const int*)d_in[7];

    const int C = in_sizes[4];  // rho has C elements

    float* out_loss = (float*)d_out;                 // first C floats
    v2f*   out_rhs  = (v2f*)((float*)d_out + C);     // next 2*C floats, 8B aligned

    const int blocks = (C + THREADS - 1) / THREADS;
    hipLaunchKernelGGL(fvm_cell_kernel, dim3(blocks), dim3(THREADS), 0, stream,
                       uv_face, p_face, flux_D, unv, rho, rhs_coef, face_area,
                       cell_face, out_loss, out_rhs, C);
}